// DecoderLayer_5377299055173
// MI455X (gfx1250) — compile-verified
//
#include <hip/hip_runtime.h>
#include <math.h>
#include <stdint.h>

typedef __attribute__((ext_vector_type(2))) float v2f;
typedef __attribute__((ext_vector_type(8))) float v8f;

#define DIMV   4096
#define NHEAD  32
#define NKV    8
#define HD     128
#define GQ     4
#define NB     64
#define INTER  14336
#define PREFIX 2048
#define SFULL  (PREFIX + 1)
#define EPSR   1e-6f
#define QKVE   ((NHEAD + 2 * NKV) * HD)   // 6144

__device__ __forceinline__ float wave_sum32(float v) {
    v += __shfl_xor(v, 16, 32);
    v += __shfl_xor(v, 8, 32);
    v += __shfl_xor(v, 4, 32);
    v += __shfl_xor(v, 2, 32);
    v += __shfl_xor(v, 1, 32);
    return v;
}
__device__ __forceinline__ float wave_max32(float v) {
    v = fmaxf(v, __shfl_xor(v, 16, 32));
    v = fmaxf(v, __shfl_xor(v, 8, 32));
    v = fmaxf(v, __shfl_xor(v, 4, 32));
    v = fmaxf(v, __shfl_xor(v, 2, 32));
    v = fmaxf(v, __shfl_xor(v, 1, 32));
    return v;
}

// --- CDNA5 async global->LDS helpers (CDNA5 ISA) -----
// LDS byte offset = low 32 bits of the generic pointer (aperture truncation,
// ISA §10.2). One b128 per lane; tracked on ASYNCcnt.
__device__ __forceinline__ void async_ld16(void* lds_dst, const void* gsrc) {
    unsigned loff = (unsigned)(uintptr_t)lds_dst;
    unsigned long long ga = (unsigned long long)(uintptr_t)gsrc;
    asm volatile("global_load_async_to_lds_b128 %0, %1, off"
                 :: "v"(loff), "v"(ga) : "memory");
}
__device__ __forceinline__ void wait_async0() {
    asm volatile("s_wait_asynccnt 0x0" ::: "memory");
}
__device__ __forceinline__ void wait_async1() {
    asm volatile("s_wait_asynccnt 0x1" ::: "memory");
}

// ---------------------------------------------------------------------------
// RMSNorm over rows of DIMV. grid = NB, block = 256 (8 waves).
// ---------------------------------------------------------------------------
__global__ __launch_bounds__(256) void rmsnorm_kernel(
    const float* __restrict__ x, const float* __restrict__ w,
    float* __restrict__ y) {
    __shared__ float part[8];
    const int row = blockIdx.x;
    const float* xr = x + (size_t)row * DIMV;
    float ss = 0.f;
    for (int i = threadIdx.x; i < DIMV; i += 256) {
        float v = xr[i];
        ss += v * v;
    }
    ss = wave_sum32(ss);
    const int lane = threadIdx.x & 31, wv = threadIdx.x >> 5;
    if (lane == 0) part[wv] = ss;
    __syncthreads();
    float tot = 0.f;
#pragma unroll
    for (int i = 0; i < 8; ++i) tot += part[i];
    const float sc = rsqrtf(tot * (1.0f / DIMV) + EPSR);
    float* yr = y + (size_t)row * DIMV;
    for (int i = threadIdx.x; i < DIMV; i += 256) yr[i] = xr[i] * sc * w[i];
}

// ---------------------------------------------------------------------------
// Per-head RMSNorm of q (heads 0..31) and k (heads 32..39) in-place in the
// qkv buffer. Rows of HD=128 floats; one wave per row.
// ---------------------------------------------------------------------------
__global__ __launch_bounds__(32) void qknorm_kernel(
    float* __restrict__ qkv, const float* __restrict__ qn,
    const float* __restrict__ kn) {
    const int r = blockIdx.x;
    const int b = r / (NHEAD + NKV);
    const int h = r % (NHEAD + NKV);     // q then k are contiguous: h*HD works
    float* p = qkv + (size_t)b * QKVE + (size_t)h * HD;
    const float* w = (h < NHEAD) ? qn : kn;
    const int lane = threadIdx.x;
    float4 v = *(const float4*)(p + lane * 4);
    float ss = v.x * v.x + v.y * v.y + v.z * v.z + v.w * v.w;
    ss = wave_sum32(ss);
    const float sc = rsqrtf(ss * (1.0f / HD) + EPSR);
    const float4 wv = *(const float4*)(w + lane * 4);
    v.x *= sc * wv.x; v.y *= sc * wv.y; v.z *= sc * wv.z; v.w *= sc * wv.w;
    *(float4*)(p + lane * 4) = v;
}

// ---------------------------------------------------------------------------
// WMMA f32 GEMM:  C[64, N] = A[64, K] * W[N, K]^T  (+ bias[n]) (+ resid[m,n])
// One wave owns a 64x16 strip (4 M-tiles, v8f acc each), K stepped by 4 via
// V_WMMA_F32_16X16X4_F32, with the 5 operand loads software-pipelined one
// K-chunk ahead so WMMA issue overlaps outstanding loads.
// block = (32,4): 4 waves = 4 consecutive n-tiles. grid.x = N/64.
// ---------------------------------------------------------------------------
__global__ __launch_bounds__(128) void gemm_wmma_kernel(
    const float* __restrict__ A, const float* __restrict__ W,
    const float* __restrict__ bias, const float* __restrict__ resid,
    float* __restrict__ C, int K, int N) {
    const int lane = threadIdx.x;
    const int kh   = lane >> 4;          // which K-half this lane provides
    const int nn   = lane & 15;
    const int ntile = blockIdx.x * 4 + threadIdx.y;
    const int n = ntile * 16 + nn;

    const float* Wr    = W + (size_t)n * K + 2 * kh;   // this lane's B row
    const float* Abase = A + (size_t)nn * K + 2 * kh;  // M-tile 0 row
    const size_t Astr  = (size_t)16 * K;

    __builtin_prefetch(Wr, 0, 0);                      // global_prefetch_b8

    v8f acc0 = {}, acc1 = {}, acc2 = {}, acc3 = {};

#define LDCH(kk, b_, a0_, a1_, a2_, a3_)                                  \
    {                                                                      \
        float2 t_;                                                         \
        t_ = *(const float2*)(Wr + (kk));             b_.x = t_.x; b_.y = t_.y;   \
        t_ = *(const float2*)(Abase + (kk));          a0_.x = t_.x; a0_.y = t_.y; \
        t_ = *(const float2*)(Abase + Astr + (kk));   a1_.x = t_.x; a1_.y = t_.y; \
        t_ = *(const float2*)(Abase + 2*Astr + (kk)); a2_.x = t_.x; a2_.y = t_.y; \
        t_ = *(const float2*)(Abase + 3*Astr + (kk)); a3_.x = t_.x; a3_.y = t_.y; \
    }
#define WMMA4(b_, a0_, a1_, a2_, a3_)                                      \
    acc0 = __builtin_amdgcn_wmma_f32_16x16x4_f32(false, a0_, false, b_,    \
                                                 (short)0, acc0, false, false); \
    acc1 = __builtin_amdgcn_wmma_f32_16x16x4_f32(false, a1_, false, b_,    \
                                                 (short)0, acc1, false, false); \
    acc2 = __builtin_amdgcn_wmma_f32_16x16x4_f32(false, a2_, false, b_,    \
                                                 (short)0, acc2, false, false); \
    acc3 = __builtin_amdgcn_wmma_f32_16x16x4_f32(false, a3_, false, b_,    \
                                                 (short)0, acc3, false, false);

    v2f bC, a0C, a1C, a2C, a3C;
    LDCH(0, bC, a0C, a1C, a2C, a3C);           // preload chunk 0
#pragma unroll 4
    for (int k = 0; k + 4 < K; k += 4) {
        v2f bN, a0N, a1N, a2N, a3N;
        LDCH(k + 4, bN, a0N, a1N, a2N, a3N);   // issue next chunk's loads
        WMMA4(bC, a0C, a1C, a2C, a3C);         // compute on current chunk
        bC = bN; a0C = a0N; a1C = a1N; a2C = a2N; a3C = a3N;
    }
    WMMA4(bC, a0C, a1C, a2C, a3C);             // drain
#undef LDCH
#undef WMMA4

    const float bn = bias ? bias[n] : 0.f;
    const int mb = 8 * kh;
#pragma unroll
    for (int r = 0; r < 8; ++r) {
        const size_t m0 = (size_t)(r + mb) * N + n;
        float v0 = acc0[r] + bn;
        float v1 = acc1[r] + bn;
        float v2 = acc2[r] + bn;
        float v3 = acc3[r] + bn;
        if (resid) {
            v0 += resid[m0];
            v1 += resid[m0 + (size_t)16 * N];
            v2 += resid[m0 + (size_t)32 * N];
            v3 += resid[m0 + (size_t)48 * N];
        }
        C[m0]                    = v0;
        C[m0 + (size_t)16 * N]   = v1;
        C[m0 + (size_t)32 * N]   = v2;
        C[m0 + (size_t)48 * N]   = v3;
    }
}

// ---------------------------------------------------------------------------
// GQA decode attention, one block per (kv-head, batch). block = 256 (8 waves).
// K and V are streamed through LDS with double-buffered
// GLOBAL_LOAD_ASYNC_TO_LDS_B128 (16B/lane), pipelined on ASYNCcnt.
// ---------------------------------------------------------------------------
__global__ __launch_bounds__(256) void attn_kernel(
    const float* __restrict__ qkv, const float* __restrict__ kc,
    const float* __restrict__ vc, float* __restrict__ out) {
    const int b = blockIdx.y, kv = blockIdx.x;
    __shared__ __align__(16) float q_s[GQ][HD];
    __shared__ __align__(16) float o_s[GQ][HD];
    __shared__ __align__(16) float tile[8][2][HD];   // per-wave double buffer
    __shared__ float sc[GQ][SFULL];
    __shared__ float red[8];
    __shared__ float stat[GQ];

    const int tid = threadIdx.x, lane = tid & 31, wv = tid >> 5;

    const float* qb = qkv + (size_t)b * QKVE + (size_t)(kv * GQ) * HD;
    for (int i = tid; i < GQ * HD; i += 256) {
        q_s[i / HD][i % HD] = qb[i];
        o_s[i / HD][i % HD] = 0.f;
    }
    __syncthreads();

    const float scale = 0.08838834764831845f;   // 1/sqrt(128)
    const float* knew = qkv + (size_t)b * QKVE + (size_t)(NHEAD + kv) * HD;
    const float* vnew = qkv + (size_t)b * QKVE + (size_t)(NHEAD + NKV + kv) * HD;

    // ---- scores: p[g][s] = q_g . k_s -----------------------------------
    {
        const float* kp0 = (0 + wv < PREFIX)
            ? (kc + (((size_t)b * PREFIX + wv) * NKV + kv) * HD) : knew;
        async_ld16(&tile[wv][0][lane * 4], kp0 + lane * 4);
        int buf = 0;
        for (int s = wv; s < SFULL; s += 8) {
            const int sn = s + 8;
            if (sn < SFULL) {
                const float* kp = (sn < PREFIX)
                    ? (kc + (((size_t)b * PREFIX + sn) * NKV + kv) * HD) : knew;
                async_ld16(&tile[wv][buf ^ 1][lane * 4], kp + lane * 4);
                wait_async1();               // current tile complete
            } else {
                wait_async0();               // drain
            }
            const float4 k4 = *(const float4*)(&tile[wv][buf][lane * 4]);
#pragma unroll
            for (int g = 0; g < GQ; ++g) {
                const float4 q4 = *(const float4*)(&q_s[g][lane * 4]);
                float d = k4.x * q4.x + k4.y * q4.y + k4.z * q4.z + k4.w * q4.w;
                d = wave_sum32(d);
                if (lane == 0) sc[g][s] = d * scale;
            }
            buf ^= 1;
        }
    }
    __syncthreads();

    // ---- softmax per group ---------------------------------------------
    for (int g = 0; g < GQ; ++g) {
        float mx = -3.4e38f;
        for (int s = tid; s < SFULL; s += 256) mx = fmaxf(mx, sc[g][s]);
        mx = wave_max32(mx);
        if (lane == 0) red[wv] = mx;
        __syncthreads();
        mx = red[0];
#pragma unroll
        for (int i = 1; i < 8; ++i) mx = fmaxf(mx, red[i]);
        float sum = 0.f;
        for (int s = tid; s < SFULL; s += 256) {
            const float e = __expf(sc[g][s] - mx);
            sc[g][s] = e;
            sum += e;
        }
        sum = wave_sum32(sum);
        __syncthreads();                 // done reading red for max
        if (lane == 0) red[wv] = sum;
        __syncthreads();
        if (tid == 0) {
            float t = 0.f;
#pragma unroll
            for (int i = 0; i < 8; ++i) t += red[i];
            stat[g] = 1.f / t;
        }
        __syncthreads();
    }

    // ---- o[g] = sum_s p[g][s] * v_s  (async-pipelined V stream) --------
    float4 acc[GQ];
#pragma unroll
    for (int g = 0; g < GQ; ++g) acc[g] = make_float4(0.f, 0.f, 0.f, 0.f);
    {
        const float* vp0 = (0 + wv < PREFIX)
            ? (vc + (((size_t)b * PREFIX + wv) * NKV + kv) * HD) : vnew;
        async_ld16(&tile[wv][0][lane * 4], vp0 + lane * 4);
        int buf = 0;
        for (int s = wv; s < SFULL; s += 8) {
            const int sn = s + 8;
            if (sn < SFULL) {
                const float* vp = (sn < PREFIX)
                    ? (vc + (((size_t)b * PREFIX + sn) * NKV + kv) * HD) : vnew;
                async_ld16(&tile[wv][buf ^ 1][lane * 4], vp + lane * 4);
                wait_async1();
            } else {
                wait_async0();
            }
            const float4 v4 = *(const float4*)(&tile[wv][buf][lane * 4]);
#pragma unroll
            for (int g = 0; g < GQ; ++g) {
                const float p = sc[g][s];
                acc[g].x += p * v4.x; acc[g].y += p * v4.y;
                acc[g].z += p * v4.z; acc[g].w += p * v4.w;
            }
            buf ^= 1;
        }
    }
#pragma unroll
    for (int g = 0; g < GQ; ++g) {        // ds_add_f32 cross-wave reduce
        atomicAdd(&o_s[g][lane * 4 + 0], acc[g].x);
        atomicAdd(&o_s[g][lane * 4 + 1], acc[g].y);
        atomicAdd(&o_s[g][lane * 4 + 2], acc[g].z);
        atomicAdd(&o_s[g][lane * 4 + 3], acc[g].w);
    }
    __syncthreads();

    float* ob = out + (size_t)b * (NHEAD * HD) + (size_t)(kv * GQ) * HD;
    for (int i = tid; i < GQ * HD; i += 256)
        ob[i] = o_s[i / HD][i % HD] * stat[i / HD];
}

// ---------------------------------------------------------------------------
// h = silu(gate) * up, elementwise (in-place on gate buffer is fine)
// ---------------------------------------------------------------------------
__global__ __launch_bounds__(256) void silu_mul_kernel(
    const float* __restrict__ g, const float* __restrict__ u,
    float* __restrict__ h, int n) {
    const int i = blockIdx.x * 256 + threadIdx.x;
    if (i < n) {
        const float x = g[i];
        h[i] = (x / (1.f + __expf(-x))) * u[i];
    }
}

// ---------------------------------------------------------------------------
extern "C" void kernel_launch(void* const* d_in, const int* in_sizes, int n_in,
                              void* d_out, int out_size, void* d_ws, size_t ws_size,
                              hipStream_t stream) {
    (void)in_sizes; (void)n_in; (void)out_size; (void)ws_size;
    const float* hidden = (const float*)d_in[0];
    const float* kc     = (const float*)d_in[1];
    const float* vc     = (const float*)d_in[2];
    const float* wqkv_w = (const float*)d_in[3];
    const float* wqkv_b = (const float*)d_in[4];
    const float* wo_w   = (const float*)d_in[5];
    const float* up_w   = (const float*)d_in[6];
    const float* gate_w = (const float*)d_in[7];
    const float* down_w = (const float*)d_in[8];
    const float* qn     = (const float*)d_in[9];
    const float* kn     = (const float*)d_in[10];
    const float* iln    = (const float*)d_in[11];
    const float* pln    = (const float*)d_in[12];
    float* out = (float*)d_out;

    float* ws      = (float*)d_ws;
    float* x_norm  = ws;                                // 64*4096
    float* qkvb    = x_norm  + (size_t)NB * DIMV;       // 64*6144
    float* attn_o  = qkvb    + (size_t)NB * QKVE;       // 64*4096
    float* hidden2 = attn_o  + (size_t)NB * DIMV;       // 64*4096
    float* x2      = hidden2 + (size_t)NB * DIMV;       // 64*4096
    float* upb     = x2      + (size_t)NB * DIMV;       // 64*14336
    float* gateb   = upb     + (size_t)NB * INTER;      // 64*14336

    const dim3 gblk(32, 4);

    // 1) input RMSNorm
    rmsnorm_kernel<<<NB, 256, 0, stream>>>(hidden, iln, x_norm);
    // 2) fused QKV projection (+bias)
    gemm_wmma_kernel<<<QKVE / 64, gblk, 0, stream>>>(x_norm, wqkv_w, wqkv_b,
                                                     nullptr, qkvb, DIMV, QKVE);
    // 3) q/k head RMSNorm in place
    qknorm_kernel<<<NB * (NHEAD + NKV), 32, 0, stream>>>(qkvb, qn, kn);
    // 4) GQA decode attention over 2048 cached + 1 new token
    attn_kernel<<<dim3(NKV, NB), 256, 0, stream>>>(qkvb, kc, vc, attn_o);
    // 5) output projection + residual
    gemm_wmma_kernel<<<DIMV / 64, gblk, 0, stream>>>(attn_o, wo_w, nullptr,
                                                     hidden, hidden2, DIMV, DIMV);
    // 6) post-attention RMSNorm
    rmsnorm_kernel<<<NB, 256, 0, stream>>>(hidden2, pln, x2);
    // 7) MLP up / gate
    gemm_wmma_kernel<<<INTER / 64, gblk, 0, stream>>>(x2, up_w, nullptr,
                                                      nullptr, upb, DIMV, INTER);
    gemm_wmma_kernel<<<INTER / 64, gblk, 0, stream>>>(x2, gate_w, nullptr,
                                                      nullptr, gateb, DIMV, INTER);
    // 8) silu(gate)*up
    silu_mul_kernel<<<(NB * INTER + 255) / 256, 256, 0, stream>>>(
        gateb, upb, gateb, NB * INTER);
    // 9) down projection + residual -> output
    gemm_wmma_kernel<<<DIMV / 64, gblk, 0, stream>>>(gateb, down_w, nullptr,
                                                     hidden2, out, INTER, DIMV);
}